// RINNController_78134045049396
// MI455X (gfx1250) — compile-verified
//
#include <hip/hip_runtime.h>

// -------- CDNA5 (gfx1250, wave32) RINN controller kernel --------------------
// base = [y|x_k] @ [Dvy|Cv]^T            (WMMA f32 16x16x4, K=80)
// w    = triangular ReLU solve            (8x cross-block WMMA + in-block sweep)
// u    = [y|x_k|w] @ [Duy|Cu|Duw]^T       (WMMA f32 16x16x4, K=208)

typedef float v2f __attribute__((ext_vector_type(2)));
typedef float v8f __attribute__((ext_vector_type(8)));

#define N_PLANT 16
#define NK 64
#define NW 128
#define NXI 80 // N_PLANT + NK
#define NU 16
#define WAVES_PER_WG 2
#define ROWS_PER_WAVE 32
#define WSTRIDE 132 // padded LDS row stride (floats): row*132 -> bank row*4

__device__ __forceinline__ v2f ld2(const float* p) {
  return *reinterpret_cast<const v2f*>(p);
}

__device__ __forceinline__ v8f wmma_f32(v2f a, v2f b, v8f c) {
  // D(16x16,f32) = A(16x4,f32) * B(4x16,f32) + C
  return __builtin_amdgcn_wmma_f32_16x16x4_f32(
      /*neg_a=*/false, a, /*neg_b=*/false, b,
      /*c_mod=*/(short)0, c, /*reuse_a=*/false, /*reuse_b=*/false);
}

// Intra-wave LDS RAW fence (DS ops are in-order per wave; this stops the
// compiler from reordering and waits out the DS pipe).
#define LDS_FENCE() asm volatile("s_wait_dscnt 0" ::: "memory")

__global__ __launch_bounds__(WAVES_PER_WG * 32)
void rinn_kernel(const float* __restrict__ xi,
                 const float* __restrict__ Cv,
                 const float* __restrict__ Dvw,
                 const float* __restrict__ Dvy,
                 const float* __restrict__ Cu,
                 const float* __restrict__ Duw,
                 const float* __restrict__ Duy,
                 float* __restrict__ out) {
  __shared__ float wbuf[WAVES_PER_WG][ROWS_PER_WAVE][WSTRIDE];

  const int tid  = threadIdx.x;
  const int lane = tid & 31;
  const int wv   = tid >> 5;
  const int lo   = lane & 15; // N column / A M-row within tile
  const int hi   = lane >> 4; // selects K pair (k, k+1) vs (k+2, k+3)
  const long b0  = (long)(blockIdx.x * WAVES_PER_WG + wv) * ROWS_PER_WAVE;

  const float* xiw0 = xi + (b0 + lo) * NXI;      // A rows, M-tile 0
  const float* xiw1 = xi + (b0 + 16 + lo) * NXI; // A rows, M-tile 1
  float (*wrow)[WSTRIDE] = wbuf[wv];

  // ---- Phase A: base, all 8 feature blocks, into LDS ----------------------
  for (int t = 0; t < 8; ++t) {
    v8f acc0 = {0.f, 0.f, 0.f, 0.f, 0.f, 0.f, 0.f, 0.f};
    v8f acc1 = {0.f, 0.f, 0.f, 0.f, 0.f, 0.f, 0.f, 0.f};
    const int n = t * 16 + lo; // output feature (B column)
#pragma unroll
    for (int kc = 0; kc < 20; ++kc) {
      const int k = kc * 4 + hi * 2; // xi column; uniform split per kc
      v2f b = (k < 16) ? ld2(&Dvy[n * N_PLANT + k])
                       : ld2(&Cv[n * NK + (k - 16)]);
      v2f a0 = ld2(&xiw0[k]);
      v2f a1 = ld2(&xiw1[k]);
      acc0 = wmma_f32(a0, b, acc0);
      acc1 = wmma_f32(a1, b, acc1);
    }
#pragma unroll
    for (int r = 0; r < 8; ++r) {
      wrow[r + 8 * hi][t * 16 + lo]      = acc0[r];
      wrow[16 + r + 8 * hi][t * 16 + lo] = acc1[r];
    }
  }
  LDS_FENCE();

  // ---- in-block sequential ReLU sweep: one lane per batch row -------------
  auto sweep = [&](int t) {
    const int c0 = t * 16;
    float v[16];
#pragma unroll
    for (int n = 0; n < 16; ++n) v[n] = wrow[lane][c0 + n];
#pragma unroll
    for (int n = 15; n >= 0; --n) {
      float wn = fmaxf(v[n], 0.0f);
      v[n] = wn;
#pragma unroll
      for (int j = 0; j < n; ++j)
        v[j] = fmaf(Dvw[(c0 + j) * NW + (c0 + n)], wn, v[j]); // uniform bcast
    }
#pragma unroll
    for (int n = 0; n < 16; ++n) wrow[lane][c0 + n] = v[n];
    LDS_FENCE();
  };

  // ---- Phase B: blocks t = 7..0 -------------------------------------------
  sweep(7); // last block has no cross-block terms
  for (int t = 6; t >= 0; --t) {
    v8f acc0, acc1;
#pragma unroll
    for (int r = 0; r < 8; ++r) { // reload base frags for block t
      acc0[r] = wrow[r + 8 * hi][t * 16 + lo];
      acc1[r] = wrow[16 + r + 8 * hi][t * 16 + lo];
    }
    const int nrow = t * 16 + lo;
    for (int s = t + 1; s < 8; ++s) { // v_t += w_s @ Dvw[t,s]^T
#pragma unroll
      for (int kc = 0; kc < 4; ++kc) {
        const int k = kc * 4 + hi * 2;
        v2f b  = ld2(&Dvw[nrow * NW + s * 16 + k]);
        v2f a0 = ld2(&wrow[lo][s * 16 + k]);
        v2f a1 = ld2(&wrow[16 + lo][s * 16 + k]);
        acc0 = wmma_f32(a0, b, acc0);
        acc1 = wmma_f32(a1, b, acc1);
      }
    }
#pragma unroll
    for (int r = 0; r < 8; ++r) {
      wrow[r + 8 * hi][t * 16 + lo]      = acc0[r];
      wrow[16 + r + 8 * hi][t * 16 + lo] = acc1[r];
    }
    LDS_FENCE();
    sweep(t);
  }

  // ---- Phase C: u = [y|x_k|w] @ [Duy|Cu|Duw]^T, K = 208 -------------------
  v8f u0 = {0.f, 0.f, 0.f, 0.f, 0.f, 0.f, 0.f, 0.f};
  v8f u1 = {0.f, 0.f, 0.f, 0.f, 0.f, 0.f, 0.f, 0.f};
#pragma unroll
  for (int kc = 0; kc < 52; ++kc) {
    const int k = kc * 4 + hi * 2;
    v2f b;
    if (k < 16)      b = ld2(&Duy[lo * N_PLANT + k]);
    else if (k < 80) b = ld2(&Cu[lo * NK + (k - 16)]);
    else             b = ld2(&Duw[lo * NW + (k - 80)]);
    v2f a0, a1;
    if (k < 80) { a0 = ld2(&xiw0[k]);               a1 = ld2(&xiw1[k]); }
    else        { a0 = ld2(&wrow[lo][k - 80]);      a1 = ld2(&wrow[16 + lo][k - 80]); }
    u0 = wmma_f32(a0, b, u0);
    u1 = wmma_f32(a1, b, u1);
  }
#pragma unroll
  for (int r = 0; r < 8; ++r) {
    out[(b0 + r + 8 * hi) * NU + lo]      = u0[r];
    out[(b0 + 16 + r + 8 * hi) * NU + lo] = u1[r];
  }
}

extern "C" void kernel_launch(void* const* d_in, const int* in_sizes, int n_in,
                              void* d_out, int out_size, void* d_ws, size_t ws_size,
                              hipStream_t stream) {
  const float* xi  = (const float*)d_in[0];
  const float* Cv  = (const float*)d_in[1];
  const float* Dvw = (const float*)d_in[2];
  const float* Dvy = (const float*)d_in[3];
  const float* Cu  = (const float*)d_in[4];
  const float* Duw = (const float*)d_in[5];
  const float* Duy = (const float*)d_in[6];
  float* out = (float*)d_out;

  const int B = in_sizes[0] / NXI; // 131072
  const int rows_per_wg = WAVES_PER_WG * ROWS_PER_WAVE;
  dim3 grid((B + rows_per_wg - 1) / rows_per_wg);
  dim3 block(WAVES_PER_WG * 32);
  rinn_kernel<<<grid, block, 0, stream>>>(xi, Cv, Dvw, Dvy, Cu, Duw, Duy, out);
}